// PPM_83210696393461
// MI455X (gfx1250) — compile-verified
//
#include <hip/hip_runtime.h>
#include <hip/hip_bf16.h>
#include <stdint.h>

// ---------------- problem constants ----------------
#define C_E      5
#define C_H      64
#define C_N      64
#define C_MB     64
#define C_NDET   16
#define C_L      8
#define C_DIA    110
#define C_V      (C_MB * C_N)          // 4096
#define C_NVOX   (C_H * C_N * C_N)     // 262144
#define C_K      (C_DIA * C_V)         // 450560
#define C_SCALE  (0.01f / 64.0f)       // CM / N
#define C_PROBE  1.0e7f

typedef float v2f __attribute__((ext_vector_type(2)));
typedef float v8f __attribute__((ext_vector_type(8)));

// ---------------- K0: zero SA accumulator ----------------
__global__ __launch_bounds__(256) void zero_sa_kernel(float* __restrict__ sa) {
    int i = blockIdx.x * 256 + threadIdx.x;
    if (i < C_L * C_V) sa[i] = 0.0f;
}

// ---------------- K1: rotate + xp overwrite + pack voxel-major ----------------
// concT layout: concT[vox*8 + e], e = 0..4 valid (pad to 8 floats -> 32B/voxel)
__global__ __launch_bounds__(256) void rotate_pack_kernel(
    const float* __restrict__ grid,     // (E,H,N,N)
    const float* __restrict__ xp,       // (E,MB,N)
    const int*   __restrict__ theta_idx,
    float* __restrict__ concT) {
    const int vox = blockIdx.x * 256 + threadIdx.x;
    if (vox >= C_NVOX) return;
    const int x = vox & 63;
    const int y = (vox >> 6) & 63;
    const int h = vox >> 12;

    float v0, v1, v2, v3, v4;
    if (h == 0) {
        // h==0 slice is replaced by xp (P_BLK=0): row m=y, col n=x
        const int o = y * 64 + x;
        v0 = xp[o];
        v1 = xp[C_V + o];
        v2 = xp[2 * C_V + o];
        v3 = xp[3 * C_V + o];
        v4 = xp[4 * C_V + o];
    } else {
        const float theta = -(float)(*theta_idx) * (6.2831853071795864f / 200.0f);
        float st, ct;
        __sincosf(theta, &st, &ct);
        const float cc = 31.5f;
        const float ys = (float)y - cc, xs = (float)x - cc;
        const float ysrc = ct * ys - st * xs + cc;
        const float xsrc = st * ys + ct * xs + cc;
        const float y0 = floorf(ysrc), x0 = floorf(xsrc);
        const float wy = ysrc - y0, wx = xsrc - x0;

        const float cw[4] = {(1.f - wy) * (1.f - wx), (1.f - wy) * wx,
                             wy * (1.f - wx),         wy * wx};
        const float cy[4] = {y0, y0, y0 + 1.f, y0 + 1.f};
        const float cx[4] = {x0, x0 + 1.f, x0, x0 + 1.f};

        v0 = v1 = v2 = v3 = v4 = 0.0f;
#pragma unroll
        for (int k = 0; k < 4; ++k) {
            const float yf = cy[k], xf = cx[k];
            const bool inb = (yf >= 0.f) && (yf <= 63.f) && (xf >= 0.f) && (xf <= 63.f);
            const float w = inb ? cw[k] : 0.0f;
            const int yi = (int)fminf(fmaxf(yf, 0.f), 63.f);
            const int xi = (int)fminf(fmaxf(xf, 0.f), 63.f);
            const int off = (h * 64 + yi) * 64 + xi;
            v0 = fmaf(w, grid[off], v0);
            v1 = fmaf(w, grid[C_NVOX + off], v1);
            v2 = fmaf(w, grid[2 * C_NVOX + off], v2);
            v3 = fmaf(w, grid[3 * C_NVOX + off], v3);
            v4 = fmaf(w, grid[4 * C_NVOX + off], v4);
        }
    }
    float* o = concT + ((size_t)vox << 3);
    *reinterpret_cast<float4*>(o) = make_float4(v0, v1, v2, v3);
    o[4] = v4;
}

// ---------------- K2: async-stage idx/len, gather-reduce S, att via WMMA ----------------
// grid: (m = 64 rows, d = 16 detectors); 64 threads, lane owns v = m*64 + tid.
__global__ __launch_bounds__(64) void fl_main_kernel(
    const int*   __restrict__ P_idx,   // (N_DET, K)
    const float* __restrict__ P_len,   // (N_DET, K)
    const float* __restrict__ FL,      // (E, L)
    const float* __restrict__ concT,   // (NVOX, 8)
    float* __restrict__ SA_acc) {      // (L, V)
    __shared__ alignas(16) int   sIdx[64 * C_DIA];
    __shared__ alignas(16) float sLen[64 * C_DIA];
    __shared__ float sS[64][9];        // stride 9 -> conflict-free strided reads

    const int m = blockIdx.x;
    const int d = blockIdx.y;
    const int tid = threadIdx.x;

    // Stage this block's idx/len chunk into LDS with the CDNA5 async-to-LDS path.
    // Global base: d*K*4 and m*7040*4 bytes are both multiples of 16 -> b128 aligned.
    const size_t base = (size_t)d * C_K + (size_t)(m * 64) * C_DIA;
    {
        const int*   gI = P_idx + base;
        const float* gL = P_len + base;
        for (int j = tid * 4; j < 64 * C_DIA; j += 256) {
            const uint32_t lI = (uint32_t)(uintptr_t)(&sIdx[j]);   // LDS byte offset
            const uint32_t lL = (uint32_t)(uintptr_t)(&sLen[j]);
            const uint64_t aI = (uint64_t)(uintptr_t)(gI + j);
            const uint64_t aL = (uint64_t)(uintptr_t)(gL + j);
            asm volatile("global_load_async_to_lds_b128 %0, %1, off"
                         :: "v"(lI), "v"(aI) : "memory");
            asm volatile("global_load_async_to_lds_b128 %0, %1, off"
                         :: "v"(lL), "v"(aL) : "memory");
        }
        asm volatile("s_wait_asynccnt 0x0" ::: "memory");
    }
    __syncthreads();

    // Gather-reduce: S[e] for this lane's v.  (stride-110 LDS reads: gcd(110,64)=2,
    // distinct banks across 32 lanes -> conflict-free)
    float a0 = 0.f, a1 = 0.f, a2 = 0.f, a3 = 0.f, a4 = 0.f;
    const int koff = tid * C_DIA;
#pragma unroll 2
    for (int i = 0; i < C_DIA; ++i) {
        const int idx = sIdx[koff + i];
        const float len = sLen[koff + i];
        const float* cp = concT + ((size_t)idx << 3);
        const float4 cv = *reinterpret_cast<const float4*>(cp);
        const float c4 = cp[4];
        a0 = fmaf(cv.x, len, a0);
        a1 = fmaf(cv.y, len, a1);
        a2 = fmaf(cv.z, len, a2);
        a3 = fmaf(cv.w, len, a3);
        a4 = fmaf(c4, len, a4);
    }
    sS[tid][0] = a0; sS[tid][1] = a1; sS[tid][2] = a2; sS[tid][3] = a3; sS[tid][4] = a4;
    __syncthreads();

    // att[l, v] = sum_e S[e,v] * FL[e,l] as (16x16x4) WMMA, K split 4 + 1.
    const int lane = tid & 31;
    const int wv   = tid >> 5;
    const int half = lane >> 4;   // 0: K=0,1 ; 1: K=2,3 (A wrap pattern)
    const int mr   = lane & 15;

    // B-matrix: B[k][n] = FL[e=k, l=n] (n<8), zero-padded; e=4 handled in B2 row K=0.
    v2f B1, B2;
    B1.x = (mr < 8) ? FL[(2 * half + 0) * C_L + mr] : 0.f;
    B1.y = (mr < 8) ? FL[(2 * half + 1) * C_L + mr] : 0.f;
    B2.x = (half == 0 && mr < 8) ? FL[4 * C_L + mr] : 0.f;
    B2.y = 0.f;

#pragma unroll
    for (int t = 0; t < 2; ++t) {
        const int rowBase = wv * 32 + t * 16;   // v tile within block
        const int r = rowBase + mr;
        v2f A1, A2;
        A1.x = sS[r][2 * half + 0];
        A1.y = sS[r][2 * half + 1];
        A2.x = (half == 0) ? sS[r][4] : 0.f;    // e=4 in K row 0 only
        A2.y = 0.f;

        v8f c = {0.f, 0.f, 0.f, 0.f, 0.f, 0.f, 0.f, 0.f};
        c = __builtin_amdgcn_wmma_f32_16x16x4_f32(false, A1, false, B1, (short)0, c, false, false);
        c = __builtin_amdgcn_wmma_f32_16x16x4_f32(false, A2, false, B2, (short)0, c, false, false);

        // D[rr + 8*half][n=mr] = att[v = rowBase + 8*half + rr][l = mr]
        if (mr < 8) {
            const int vloc = m * 64 + rowBase + half * 8;
#pragma unroll
            for (int rr = 0; rr < 8; ++rr) {
                atomicAdd(&SA_acc[mr * C_V + vloc + rr], __expf(-c[rr]));
            }
        }
    }
}

// ---------------- K3: attenuation prefix, transmission, final reductions ----------------
__global__ __launch_bounds__(64) void fl_finalize_kernel(
    const float* __restrict__ xp,         // (E,MB,N)
    const float* __restrict__ probe_att,  // (E)
    const float* __restrict__ dfl,        // (L)
    const float* __restrict__ SA_acc,     // (L,V) summed over d
    float* __restrict__ out) {            // [0..511] fl_signal (L,MB), [512..575] transmission
    __shared__ float sh[64];
    __shared__ float red[64];
    const int m = blockIdx.x;
    const int n = threadIdx.x;
    const int v = m * 64 + n;

    float lac = 0.f;
#pragma unroll
    for (int e = 0; e < C_E; ++e) lac = fmaf(xp[e * C_V + v], probe_att[e], lac);
    sh[n] = lac;
    __syncthreads();

    float pre = 0.f, total = 0.f;
    for (int j = 0; j < 64; ++j) {
        const float t = sh[j];
        total += t;
        if (j < n) pre += t;
    }
    const float probe_after = C_PROBE * __expf(-pre * C_SCALE);
    if (n == 0) out[512 + m] = C_PROBE * __expf(-total * C_SCALE);

    const int l2e[C_L] = {0, 1, 2, 2, 3, 3, 4, 4};
#pragma unroll
    for (int l = 0; l < C_L; ++l) {
        const float flm = dfl[l] * xp[l2e[l] * C_V + v];
        const float sa = SA_acc[l * C_V + v] * (1.0f / 16.0f);
        red[n] = probe_after * flm * sa;
        __syncthreads();
        if (n == 0) {
            float s = 0.f;
            for (int j = 0; j < 64; ++j) s += red[j];
            out[l * 64 + m] = s;
        }
        __syncthreads();
    }
}

// ---------------- launch ----------------
extern "C" void kernel_launch(void* const* d_in, const int* in_sizes, int n_in,
                              void* d_out, int out_size, void* d_ws, size_t ws_size,
                              hipStream_t stream) {
    // setup_inputs() dict order
    const float* grid      = (const float*)d_in[0]; // grid_concentration
    const float* xp        = (const float*)d_in[1]; // xp
    const float* P_len     = (const float*)d_in[2]; // P_len
    const float* dfl       = (const float*)d_in[3]; // detected_fl_unit
    const float* FL        = (const float*)d_in[4]; // FL_line_attCS
    const float* probe_att = (const float*)d_in[5]; // probe_attCS
    const int*   P_idx     = (const int*)d_in[6];   // P_idx
    const int*   theta_idx = (const int*)d_in[7];   // this_theta_idx
    float* out = (float*)d_out;

    float* concT  = (float*)d_ws;                 // NVOX * 8 floats = 8 MB
    float* SA_acc = concT + (size_t)C_NVOX * 8;   // L * V floats = 128 KB

    zero_sa_kernel<<<(C_L * C_V + 255) / 256, 256, 0, stream>>>(SA_acc);
    rotate_pack_kernel<<<C_NVOX / 256, 256, 0, stream>>>(grid, xp, theta_idx, concT);
    fl_main_kernel<<<dim3(64, 16), 64, 0, stream>>>(P_idx, P_len, FL, concT, SA_acc);
    fl_finalize_kernel<<<64, 64, 0, stream>>>(xp, probe_att, dfl, SA_acc, out);
}